// SuperPointMatching_33174327394662
// MI455X (gfx1250) — compile-verified
//
#include <hip/hip_runtime.h>
#include <stdint.h>

// Problem constants (fixed by the reference harness).
#define B_ 8
#define N_ 2048
#define M_ 2048
#define K_ 256
#define NM_ (N_ * M_)

typedef __attribute__((ext_vector_type(2))) float v2f;
typedef __attribute__((ext_vector_type(8))) float v8f;

// ---------------------------------------------------------------------------
// Kernel 1: row sums via V_WMMA_F32_16X16X4_F32  (rowsum = C_tile x ones)
// A-matrix 16x4 f32 layout (ISA 7.12.2): lanes 0-15 hold K=0,1 (vgpr0,1) for
// M=lane; lanes 16-31 hold K=2,3 for M=lane-16. B = all-ones 4x16 (layout
// irrelevant since every element is 1.0). D: vgpr v, lanes 0-15 -> M=v,
// lanes 16-31 -> M=8+v; every column N holds the same row sum.
// ---------------------------------------------------------------------------
__global__ __launch_bounds__(32)
void rowsum_wmma_kernel(const float* __restrict__ c, float* __restrict__ row_sums) {
    const int b    = blockIdx.y;
    const int n0   = blockIdx.x << 4;          // 16 rows per wave
    const int lane = threadIdx.x;              // wave32
    const int row  = lane & 15;
    const int koff = (lane >> 4) << 1;         // lanes 16-31 handle K=2,3
    const float* base = c + (((size_t)b * N_ + n0 + row) * M_ + koff);

    v8f a0 = {}, a1 = {}, a2 = {}, a3 = {};
    const v2f ones = {1.0f, 1.0f};
    for (int k = 0; k < M_; k += 16) {
        v2f x0 = *(const v2f*)(base + k);
        v2f x1 = *(const v2f*)(base + k + 4);
        v2f x2 = *(const v2f*)(base + k + 8);
        v2f x3 = *(const v2f*)(base + k + 12);
        a0 = __builtin_amdgcn_wmma_f32_16x16x4_f32(false, x0, false, ones, (short)0, a0, false, false);
        a1 = __builtin_amdgcn_wmma_f32_16x16x4_f32(false, x1, false, ones, (short)0, a1, false, false);
        a2 = __builtin_amdgcn_wmma_f32_16x16x4_f32(false, x2, false, ones, (short)0, a2, false, false);
        a3 = __builtin_amdgcn_wmma_f32_16x16x4_f32(false, x3, false, ones, (short)0, a3, false, false);
    }
    v8f acc = a0 + a1 + a2 + a3;

    if (lane == 0) {
#pragma unroll
        for (int i = 0; i < 8; ++i) row_sums[(size_t)b * N_ + n0 + i] = acc[i];
    } else if (lane == 16) {
#pragma unroll
        for (int i = 0; i < 8; ++i) row_sums[(size_t)b * N_ + n0 + 8 + i] = acc[i];
    }
}

// ---------------------------------------------------------------------------
// Kernel 2: column sums, fully coalesced (thread t owns column m).
// ---------------------------------------------------------------------------
__global__ __launch_bounds__(256)
void colsum_kernel(const float* __restrict__ c, float* __restrict__ col_sums) {
    const int b = blockIdx.y;
    const int m = blockIdx.x * 256 + threadIdx.x;
    const float* base = c + ((size_t)b * NM_ + m);
    float s = 0.0f;
    for (int n = 0; n < N_; ++n) s += base[(size_t)n * M_];
    col_sums[b * M_ + m] = s;
}

// Bit-identical score across all kernels (plain IEEE ops, no contraction
// possible across div/mul boundaries).
__device__ __forceinline__ float score_of(float cv, float rs, float cs) {
    return (cv / rs) * (cv / cs);
}

// state[b*8 + i]: 0=prefix mask, 1=prefix val (-> threshold T), 2=k remaining
// within prefix, 3=count strictly greater than T, 4=collect >T counter,
// 5=collect ==T counter.
__global__ void init_kernel(unsigned* __restrict__ hist, unsigned* __restrict__ state) {
    const int b = blockIdx.x, t = threadIdx.x;
    hist[b * 256 + t] = 0u;
    if (t < 8) state[b * 8 + t] = (t == 2) ? (unsigned)K_ : 0u;
}

// ---------------------------------------------------------------------------
// Kernel 3: radix-select histogram pass (8-bit digit of the fp32 score bits).
// Scores are >= 0 so unsigned bit order == value order.
// ---------------------------------------------------------------------------
__global__ __launch_bounds__(256)
void hist_kernel(const float* __restrict__ c, const float* __restrict__ row_sums,
                 const float* __restrict__ col_sums, unsigned* __restrict__ hist,
                 const unsigned* __restrict__ state, int pass) {
    __shared__ unsigned lh[256];
    const int b = blockIdx.y, t = threadIdx.x;
    lh[t] = 0u;
    __syncthreads();

    const unsigned mask = state[b * 8 + 0];
    const unsigned val  = state[b * 8 + 1];
    const int sh = 24 - 8 * pass;
    const size_t baseIdx = (size_t)b * NM_;
    const int stride = gridDim.x * 256;
    for (int e = blockIdx.x * 256 + t; e < NM_; e += stride) {
        const int n = e >> 11;            // / M_
        const int m = e & (M_ - 1);       // % M_
        const float s = score_of(c[baseIdx + e], row_sums[b * N_ + n], col_sums[b * M_ + m]);
        const unsigned u = __float_as_uint(s);
        if ((u & mask) == val) atomicAdd(&lh[(u >> sh) & 255u], 1u);
    }
    __syncthreads();
    if (lh[t]) atomicAdd(&hist[b * 256 + t], lh[t]);
}

// Kernel 4: pick the digit containing the Kth largest; re-zero the histogram.
__global__ void scan_kernel(unsigned* __restrict__ hist, unsigned* __restrict__ state, int pass) {
    const int b = blockIdx.x;
    if (threadIdx.x != 0) return;
    unsigned* st = state + b * 8;
    unsigned* h  = hist + b * 256;
    const unsigned kneed = st[2];
    unsigned acc = 0;
    int chosen = 0;
    for (int bin = 255; bin >= 0; --bin) {
        const unsigned cb = h[bin];
        if (acc + cb >= kneed) { chosen = bin; break; }
        acc += cb;
    }
    const int sh = 24 - 8 * pass;
    st[0] |= 0xFFu << sh;
    st[1] |= ((unsigned)chosen) << sh;
    st[2]  = kneed - acc;
    if (pass == 3) st[3] = (unsigned)K_ - st[2];
    for (int i = 0; i < 256; ++i) h[i] = 0u;
}

// ---------------------------------------------------------------------------
// Kernel 5: gather the K winners (all >T, plus the needed ==T ties).
// ---------------------------------------------------------------------------
__global__ __launch_bounds__(256)
void collect_kernel(const float* __restrict__ c, const float* __restrict__ row_sums,
                    const float* __restrict__ col_sums, unsigned* __restrict__ state,
                    unsigned* __restrict__ cand_idx, float* __restrict__ cand_score) {
    const int b = blockIdx.y, t = threadIdx.x;
    unsigned* st = state + b * 8;
    const unsigned T    = st[1];
    const unsigned kEq  = st[2];
    const unsigned cntG = st[3];
    const size_t baseIdx = (size_t)b * NM_;
    const int stride = gridDim.x * 256;
    for (int e = blockIdx.x * 256 + t; e < NM_; e += stride) {
        const int n = e >> 11;
        const int m = e & (M_ - 1);
        const float s = score_of(c[baseIdx + e], row_sums[b * N_ + n], col_sums[b * M_ + m]);
        const unsigned u = __float_as_uint(s);
        if (u > T) {
            const unsigned p = atomicAdd(&st[4], 1u);
            if (p < (unsigned)K_) { cand_idx[b * K_ + p] = (unsigned)e; cand_score[b * K_ + p] = s; }
        } else if (u == T) {
            const unsigned q = atomicAdd(&st[5], 1u);
            if (q < kEq) { cand_idx[b * K_ + cntG + q] = (unsigned)e; cand_score[b * K_ + cntG + q] = s; }
        }
    }
}

// ---------------------------------------------------------------------------
// Kernel 6: per-batch 256-element bitonic sort (descending score, ties by
// ascending flat index, matching jax.lax.top_k), then emit the three output
// segments: ref_idx | src_idx | scores (flat, batch-major).
// ---------------------------------------------------------------------------
__global__ __launch_bounds__(256)
void sort_out_kernel(const unsigned* __restrict__ cand_idx, const float* __restrict__ cand_score,
                     float* __restrict__ out) {
    __shared__ float    ss[256];
    __shared__ unsigned si[256];
    const int b = blockIdx.x, t = threadIdx.x;
    ss[t] = cand_score[b * K_ + t];
    si[t] = cand_idx[b * K_ + t];
    __syncthreads();

    for (int ksz = 2; ksz <= 256; ksz <<= 1) {
        for (int j = ksz >> 1; j > 0; j >>= 1) {
            const int ixj = t ^ j;
            if (ixj > t) {
                const float s0 = ss[t], s1 = ss[ixj];
                const unsigned i0 = si[t], i1 = si[ixj];
                // "t comes after ixj" in the desired descending order:
                const bool after_t = (s0 < s1) || (s0 == s1 && i0 > i1);
                const bool doSwap = ((t & ksz) == 0) ? after_t : !after_t;
                if (doSwap) { ss[t] = s1; ss[ixj] = s0; si[t] = i1; si[ixj] = i0; }
            }
            __syncthreads();
        }
    }

    const unsigned fi = si[t];
    out[0 * (B_ * K_) + b * K_ + t] = (float)(fi >> 11);       // flat // M
    out[1 * (B_ * K_) + b * K_ + t] = (float)(fi & (M_ - 1));  // flat % M
    out[2 * (B_ * K_) + b * K_ + t] = ss[t];
}

// ---------------------------------------------------------------------------
extern "C" void kernel_launch(void* const* d_in, const int* in_sizes, int n_in,
                              void* d_out, int out_size, void* d_ws, size_t ws_size,
                              hipStream_t stream) {
    (void)in_sizes; (void)n_in; (void)out_size; (void)ws_size;
    const float* c = (const float*)d_in[2];  // c_matrix [B,N,M]; other inputs unused

    char* ws = (char*)d_ws;
    float*    row_sums   = (float*)ws;    ws += (size_t)B_ * N_ * sizeof(float);
    float*    col_sums   = (float*)ws;    ws += (size_t)B_ * M_ * sizeof(float);
    unsigned* hist       = (unsigned*)ws; ws += (size_t)B_ * 256 * sizeof(unsigned);
    unsigned* state      = (unsigned*)ws; ws += (size_t)B_ * 8 * sizeof(unsigned);
    unsigned* cand_idx   = (unsigned*)ws; ws += (size_t)B_ * K_ * sizeof(unsigned);
    float*    cand_score = (float*)ws;
    float*    out        = (float*)d_out;

    dim3 gRow(N_ / 16, B_);
    rowsum_wmma_kernel<<<gRow, 32, 0, stream>>>(c, row_sums);

    dim3 gCol(M_ / 256, B_);
    colsum_kernel<<<gCol, 256, 0, stream>>>(c, col_sums);

    init_kernel<<<B_, 256, 0, stream>>>(hist, state);

    dim3 gSel(512, B_);
    for (int p = 0; p < 4; ++p) {
        hist_kernel<<<gSel, 256, 0, stream>>>(c, row_sums, col_sums, hist, state, p);
        scan_kernel<<<B_, 32, 0, stream>>>(hist, state, p);
    }

    collect_kernel<<<gSel, 256, 0, stream>>>(c, row_sums, col_sums, state, cand_idx, cand_score);
    sort_out_kernel<<<B_, 256, 0, stream>>>(cand_idx, cand_score, out);
}